// GAT_89077621719519
// MI455X (gfx1250) — compile-verified
//
#include <hip/hip_runtime.h>
#include <cmath>

#define HCID 128
#define NEG_SLOPE 0.2f
#define BN_EPS 1e-5f

typedef __attribute__((ext_vector_type(2))) float v2f;
typedef __attribute__((ext_vector_type(8))) float v8f;

// ---------- helpers ----------
__device__ __forceinline__ unsigned fenc(float x) {
    unsigned u = __float_as_uint(x);
    return (u & 0x80000000u) ? ~u : (u | 0x80000000u);
}
__device__ __forceinline__ float fdec(unsigned e) {
    unsigned u = (e & 0x80000000u) ? (e & 0x7FFFFFFFu) : ~e;
    return __uint_as_float(u);
}
__device__ __forceinline__ float lrelu(float v) { return v < 0.f ? NEG_SLOPE * v : v; }

// ---------- WMMA GEMM: Y[nrows,NCOLS] = X[nrows,K] @ W[K,NCOLS] (+bias) ----------
// One wave per 16-row tile.  f32 16x16x4 WMMA.
// A layout (32-bit A 16x4): lanes 0-15 row M=l hold K=k0,k0+1 (V0,V1); lanes 16-31 hold K=k0+2,k0+3.
// B layout (32-bit B 4x16): lanes 0-15 col N=l hold K=k0,k0+1; lanes 16-31 hold K=k0+2,k0+3.
// C/D: VGPR i: lanes0-15 -> M=i, lanes16-31 -> M=8+i.
template <int NCOLS>
__global__ __launch_bounds__(32) void gemm_wmma(const float* __restrict__ X,
                                                const float* __restrict__ W,
                                                float* __restrict__ Y,
                                                int nrows, int K,
                                                const float* __restrict__ bias) {
    constexpr int NT = NCOLS / 16;
    int row0 = blockIdx.x * 16;
    int lane = threadIdx.x & 31;
    int half = lane >> 4;
    int l = lane & 15;

    v8f acc[NT];
#pragma unroll
    for (int t = 0; t < NT; ++t)
#pragma unroll
        for (int i = 0; i < 8; ++i) acc[t][i] = 0.f;

    int arow = row0 + l;
    if (arow >= nrows) arow = nrows - 1;

    for (int k0 = 0; k0 < K; k0 += 4) {
        const float* xr = X + (size_t)arow * K + (k0 + 2 * half);
        v2f a;
        a.x = xr[0];
        a.y = xr[1];
        const float* w0 = W + (size_t)(k0 + 2 * half) * NCOLS + l;
#pragma unroll
        for (int t = 0; t < NT; ++t) {
            v2f b;
            b.x = w0[t * 16];
            b.y = w0[NCOLS + t * 16];
            acc[t] = __builtin_amdgcn_wmma_f32_16x16x4_f32(false, a, false, b,
                                                           (short)0, acc[t],
                                                           false, false);
        }
    }
#pragma unroll
    for (int t = 0; t < NT; ++t) {
#pragma unroll
        for (int i = 0; i < 8; ++i) {
            int row = row0 + half * 8 + i;
            if (row < nrows) {
                int col = t * 16 + l;
                float y = acc[t][i];
                if (bias) y += bias[col];
                Y[(size_t)row * NCOLS + col] = y;
            }
        }
    }
}

// ---------- ce[h] = sum_c We[h*32+c] * ae[h*32+c] ----------
__global__ void compute_ce(const float* __restrict__ We, const float* __restrict__ ae,
                           float* __restrict__ ce) {
    int h = threadIdx.x;
    if (h < 4) {
        float s = 0.f;
        for (int c = 0; c < 32; ++c) s += We[h * 32 + c] * ae[h * 32 + c];
        ce[h] = s;
    }
}

// ---------- per-node attention scores asrc/adst [N*4] ----------
__global__ void attn_scores(const float* __restrict__ xl, const float* __restrict__ a_src,
                            const float* __restrict__ a_dst, float* __restrict__ asrc,
                            float* __restrict__ adst, int n) {
    int i = blockIdx.x * blockDim.x + threadIdx.x;
    if (i >= n * 4) return;
    int node = i >> 2, h = i & 3;
    const float* xr = xl + (size_t)node * HCID + h * 32;
    float s1 = 0.f, s2 = 0.f;
#pragma unroll
    for (int c = 0; c < 32; ++c) {
        float v = xr[c];
        s1 += v * a_src[h * 32 + c];
        s2 += v * a_dst[h * 32 + c];
    }
    asrc[i] = s1;
    adst[i] = s2;
}

// ---------- segment max over destinations (edges) ----------
__global__ void edge_max(const int* __restrict__ src, const int* __restrict__ dst,
                         const float* __restrict__ ew, const float* __restrict__ asrc,
                         const float* __restrict__ adst, const float* __restrict__ ce,
                         unsigned* __restrict__ menc, int E) {
    int e = blockIdx.x * blockDim.x + threadIdx.x;
    if (e >= E) return;
    int s = src[e], d = dst[e];
    float w = ew[e];
#pragma unroll
    for (int h = 0; h < 4; ++h) {
        float v = lrelu(asrc[s * 4 + h] + adst[d * 4 + h] + w * ce[h]);
        atomicMax(&menc[d * 4 + h], fenc(v));
    }
}

__global__ void self_max(const float* __restrict__ asrc, const float* __restrict__ adst,
                         unsigned* __restrict__ menc, int n4) {
    int i = blockIdx.x * blockDim.x + threadIdx.x;
    if (i >= n4) return;
    float v = lrelu(asrc[i] + adst[i]);
    atomicMax(&menc[i], fenc(v));
}

// ---------- segment sum of exp(logit - m) ----------
__global__ void edge_denom(const int* __restrict__ src, const int* __restrict__ dst,
                           const float* __restrict__ ew, const float* __restrict__ asrc,
                           const float* __restrict__ adst, const float* __restrict__ ce,
                           const unsigned* __restrict__ menc, float* __restrict__ denom,
                           int E) {
    int e = blockIdx.x * blockDim.x + threadIdx.x;
    if (e >= E) return;
    int s = src[e], d = dst[e];
    float w = ew[e];
#pragma unroll
    for (int h = 0; h < 4; ++h) {
        float v = lrelu(asrc[s * 4 + h] + adst[d * 4 + h] + w * ce[h]);
        atomicAdd(&denom[d * 4 + h], expf(v - fdec(menc[d * 4 + h])));
    }
}

__global__ void self_denom(const float* __restrict__ asrc, const float* __restrict__ adst,
                           const unsigned* __restrict__ menc, float* __restrict__ denom,
                           int n4) {
    int i = blockIdx.x * blockDim.x + threadIdx.x;
    if (i >= n4) return;
    float v = lrelu(asrc[i] + adst[i]);
    denom[i] += expf(v - fdec(menc[i]));  // serialized after edge_denom on stream
}

// ---------- message scatter: out[d,:] += alpha * xl[s,:]  (thread per edge-channel) ----------
__global__ void edge_msg(const int* __restrict__ src, const int* __restrict__ dst,
                         const float* __restrict__ ew, const float* __restrict__ asrc,
                         const float* __restrict__ adst, const float* __restrict__ ce,
                         const unsigned* __restrict__ menc, const float* __restrict__ denom,
                         const float* __restrict__ xl, float* __restrict__ out,
                         long long total) {
    long long g = (long long)blockIdx.x * blockDim.x + threadIdx.x;
    if (g >= total) return;
    int e = (int)(g >> 7);
    int c = (int)(g & 127);
    int h = c >> 5;
    int s = src[e], d = dst[e];
    float v = lrelu(asrc[s * 4 + h] + adst[d * 4 + h] + ew[e] * ce[h]);
    float alpha = expf(v - fdec(menc[d * 4 + h])) / denom[d * 4 + h];
    atomicAdd(&out[(size_t)d * HCID + c], alpha * xl[(size_t)s * HCID + c]);
}

// ---------- self-loop message + bias ----------
__global__ void self_msg_bias(const float* __restrict__ asrc, const float* __restrict__ adst,
                              const unsigned* __restrict__ menc, const float* __restrict__ denom,
                              const float* __restrict__ xl, const float* __restrict__ b,
                              float* __restrict__ out, long long total) {
    long long g = (long long)blockIdx.x * blockDim.x + threadIdx.x;
    if (g >= total) return;
    int node = (int)(g >> 7);
    int c = (int)(g & 127);
    int h = c >> 5;
    float v = lrelu(asrc[node * 4 + h] + adst[node * 4 + h]);
    float alpha = expf(v - fdec(menc[node * 4 + h])) / denom[node * 4 + h];
    out[g] += alpha * xl[g] + b[c];
}

// ---------- residual add ----------
__global__ void add_inplace(float* __restrict__ a, const float* __restrict__ b,
                            long long total) {
    long long g = (long long)blockIdx.x * blockDim.x + threadIdx.x;
    if (g < total) a[g] += b[g];
}

// ---------- batch-norm stats (sums[0..127]=sum, sums[128..255]=sumsq) ----------
__global__ __launch_bounds__(128) void bn_stats(const float* __restrict__ x,
                                                float* __restrict__ sums, int n) {
    int c = threadIdx.x;
    int r0 = blockIdx.x * 256;
    float s = 0.f, sq = 0.f;
    int rend = r0 + 256;
    if (rend > n) rend = n;
    for (int r = r0; r < rend; ++r) {
        float v = x[(size_t)r * HCID + c];
        s += v;
        sq += v * v;
    }
    atomicAdd(&sums[c], s);
    atomicAdd(&sums[128 + c], sq);
}

// ---------- batch-norm apply + ReLU ----------
__global__ void bn_apply(float* __restrict__ x, const float* __restrict__ sums,
                         const float* __restrict__ gam, const float* __restrict__ beta,
                         int n, long long total) {
    long long g = (long long)blockIdx.x * blockDim.x + threadIdx.x;
    if (g >= total) return;
    int c = (int)(g & 127);
    float mean = sums[c] / (float)n;
    float var = sums[128 + c] / (float)n - mean * mean;
    float y = (x[g] - mean) * rsqrtf(var + BN_EPS) * gam[c] + beta[c];
    x[g] = y > 0.f ? y : 0.f;
}

// ==================================================================
extern "C" void kernel_launch(void* const* d_in, const int* in_sizes, int n_in,
                              void* d_out, int out_size, void* d_ws, size_t ws_size,
                              hipStream_t stream) {
    const float* x = (const float*)d_in[0];
    const int* src = (const int*)d_in[1];
    const int* dst = (const int*)d_in[2];
    const float* ew = (const float*)d_in[3];

    struct P {
        const float *W, *as, *ad, *We, *ae, *b, *g, *beta;
    };
    auto getP = [&](int base) {
        P p;
        p.W = (const float*)d_in[base + 0];
        p.as = (const float*)d_in[base + 1];
        p.ad = (const float*)d_in[base + 2];
        p.We = (const float*)d_in[base + 3];
        p.ae = (const float*)d_in[base + 4];
        p.b = (const float*)d_in[base + 5];
        p.g = (const float*)d_in[base + 6];
        p.beta = (const float*)d_in[base + 7];
        return p;
    };
    P p0 = getP(4), p1 = getP(12), p2 = getP(20);
    const float* Wf = (const float*)d_in[28];
    const float* bf = (const float*)d_in[29];

    const int E = in_sizes[1];
    const int N = in_sizes[0] / 20;
    const size_t NHC = (size_t)N * HCID;
    const long long totE = (long long)E * HCID;
    const long long totN = (long long)N * HCID;

    float* ws = (float*)d_ws;
    float* xl = ws;
    float* outA = xl + NHC;    // conv0 out -> x0 (residual)
    float* outB = outA + NHC;  // conv1 out -> h; reused as conv2 out
    float* asrc = outB + NHC;
    float* adst = asrc + (size_t)N * 4;
    unsigned* menc = (unsigned*)(adst + (size_t)N * 4);
    float* denom = (float*)(menc + (size_t)N * 4);
    float* ce = denom + (size_t)N * 4;
    float* sums = ce + 8;

    auto conv = [&](const float* xin, int K, const P& p, float* out) {
        gemm_wmma<HCID><<<(N + 15) / 16, 32, 0, stream>>>(xin, p.W, xl, N, K, nullptr);
        compute_ce<<<1, 4, 0, stream>>>(p.We, p.ae, ce);
        attn_scores<<<(N * 4 + 255) / 256, 256, 0, stream>>>(xl, p.as, p.ad, asrc, adst, N);
        hipMemsetAsync(menc, 0, (size_t)N * 4 * sizeof(unsigned), stream);
        hipMemsetAsync(denom, 0, (size_t)N * 4 * sizeof(float), stream);
        hipMemsetAsync(out, 0, NHC * sizeof(float), stream);
        edge_max<<<(E + 255) / 256, 256, 0, stream>>>(src, dst, ew, asrc, adst, ce, menc, E);
        self_max<<<(N * 4 + 255) / 256, 256, 0, stream>>>(asrc, adst, menc, N * 4);
        edge_denom<<<(E + 255) / 256, 256, 0, stream>>>(src, dst, ew, asrc, adst, ce, menc,
                                                        denom, E);
        self_denom<<<(N * 4 + 255) / 256, 256, 0, stream>>>(asrc, adst, menc, denom, N * 4);
        edge_msg<<<(unsigned)((totE + 255) / 256), 256, 0, stream>>>(
            src, dst, ew, asrc, adst, ce, menc, denom, xl, out, totE);
        self_msg_bias<<<(unsigned)((totN + 255) / 256), 256, 0, stream>>>(
            asrc, adst, menc, denom, xl, p.b, out, totN);
    };
    auto bn = [&](float* buf, const float* g, const float* beta) {
        hipMemsetAsync(sums, 0, 256 * sizeof(float), stream);
        bn_stats<<<(N + 255) / 256, 128, 0, stream>>>(buf, sums, N);
        bn_apply<<<(unsigned)((totN + 255) / 256), 256, 0, stream>>>(buf, sums, g, beta, N,
                                                                     totN);
    };

    // layer 0: x0 = relu(bn(conv0(x)))
    conv(x, 20, p0, outA);
    bn(outA, p0.g, p0.beta);
    // layer 1: h = relu(bn(conv1(x)))
    conv(x, 20, p1, outB);
    bn(outB, p1.g, p1.beta);
    // layer 2: h = relu(bn(conv2(h) + x0))   (gemm reads outB into xl before outB is zeroed)
    conv(outB, HCID, p2, outB);
    add_inplace<<<(unsigned)((totN + 255) / 256), 256, 0, stream>>>(outB, outA, totN);
    bn(outB, p2.g, p2.beta);
    // fc: out = h @ Wf + bf
    gemm_wmma<32><<<(N + 15) / 16, 32, 0, stream>>>(outB, Wf, (float*)d_out, N, HCID, bf);
}